// FlowGNN_45835890983447
// MI455X (gfx1250) — compile-verified
//
#include <hip/hip_runtime.h>
#include <hip/hip_bf16.h>
#include <float.h>
#include <stdint.h>

typedef float v2f __attribute__((ext_vector_type(2)));
typedef float v8f __attribute__((ext_vector_type(8)));

#define CH 128           // feature width (C == H == 128)
#define NEG_SLOPE 0.01f
#define NUM_GRAPHS 64

// ---------- helpers ----------
__device__ __forceinline__ unsigned f2ord(float f) {
    unsigned u = __float_as_uint(f);
    return (u & 0x80000000u) ? ~u : (u | 0x80000000u);
}
__device__ __forceinline__ float ord2f(unsigned u) {
    return __uint_as_float((u & 0x80000000u) ? (u & 0x7FFFFFFFu) : ~u);
}

// ---------- init kernels (graph-capture safe zeroing) ----------
__global__ void zero_f32(float* __restrict__ p, size_t n) {
    size_t i = (size_t)blockIdx.x * blockDim.x + threadIdx.x;
    if (i < n) p[i] = 0.0f;
}
__global__ void init_pooled(unsigned* __restrict__ p, int n) {
    int i = blockIdx.x * blockDim.x + threadIdx.x;
    if (i < n) p[i] = 0x007FFFFFu;   // f2ord(-inf)
}

// ---------- degree: deg[dst[e]] += 1 ----------
__global__ __launch_bounds__(256) void degree_kernel(const int* __restrict__ dst,
                                                     float* __restrict__ deg, int E) {
    int e = blockIdx.x * blockDim.x + threadIdx.x;
    if (e < E) atomicAdd(&deg[dst[e]], 1.0f);
}

// ---------- edge scatter: agg[dst[e], :] += X[src[e], :] ----------
__global__ __launch_bounds__(256) void scatter_add(const float* __restrict__ X,
                                                   const int* __restrict__ src,
                                                   const int* __restrict__ dst,
                                                   float* __restrict__ agg, int E) {
    long long t = (long long)blockIdx.x * blockDim.x + threadIdx.x;
    int e = (int)(t >> 5);
    if (e >= E) return;
    int c4 = ((int)t & 31) << 2;
    int s = src[e], d = dst[e];
    const float4 v = *reinterpret_cast<const float4*>(X + (size_t)s * CH + c4);
    float* p = agg + (size_t)d * CH + c4;
    atomicAdd(p + 0, v.x);
    atomicAdd(p + 1, v.y);
    atomicAdd(p + 2, v.z);
    atomicAdd(p + 3, v.w);
}

// ---------- fused SAGE layer: Y = leaky( (agg/deg)@Wl + X@Wr + b ) ----------
// block = 256 threads = 8 waves; block covers 16 rows x 128 cols;
// wave w computes the 16x16 tile at columns [16w, 16w+16).
// Per-wave: 2 * (128/4) = 64 x v_wmma_f32_16x16x4_f32.
__global__ __launch_bounds__(256) void sage_gemm(const float* __restrict__ X,
                                                 const float* __restrict__ agg,
                                                 const float* __restrict__ deg,
                                                 const float* __restrict__ Wl,
                                                 const float* __restrict__ Wr,
                                                 const float* __restrict__ bias,
                                                 float* __restrict__ Y, int N) {
    __shared__ float mean_s[16][132];   // 132 pad: row stride 132%64=4 -> conflict-free
    __shared__ float x_s[16][132];

    const int row0 = blockIdx.x * 16;
    const int tid = threadIdx.x;
    const bool full = (row0 + 16 <= N);

    if (full) {
        // thread -> 8 contiguous channels of one row (32B)
        const int r = tid >> 4;
        const int c = (tid & 15) * 8;
        const size_t gbase = (size_t)(row0 + r) * CH + c;

        // x tile: async global -> LDS copy (ASYNCcnt); IOFFSET applies to both
        // the LDS and the global address, so offset:16 moves the second B128.
        {
            unsigned lds = (unsigned)(uintptr_t)(&x_s[r][c]);   // low 32b = LDS addr
            unsigned long long ga = (unsigned long long)(uintptr_t)(X + gbase);
            asm volatile(
                "global_load_async_to_lds_b128 %0, %1, off\n\t"
                "global_load_async_to_lds_b128 %0, %1, off offset:16"
                :: "v"(lds), "v"(ga) : "memory");
        }

        // mean tile: needs VALU (divide), overlap it with the async copy
        {
            const float inv = 1.0f / fmaxf(deg[row0 + r], 1.0f);
            const float4 a0 = *reinterpret_cast<const float4*>(agg + gbase);
            const float4 a1 = *reinterpret_cast<const float4*>(agg + gbase + 4);
            mean_s[r][c + 0] = a0.x * inv;  mean_s[r][c + 1] = a0.y * inv;
            mean_s[r][c + 2] = a0.z * inv;  mean_s[r][c + 3] = a0.w * inv;
            mean_s[r][c + 4] = a1.x * inv;  mean_s[r][c + 5] = a1.y * inv;
            mean_s[r][c + 6] = a1.z * inv;  mean_s[r][c + 7] = a1.w * inv;
        }
        asm volatile("s_wait_asynccnt 0x0" ::: "memory");
    } else {
        // guarded tail path
        for (int i = tid; i < 16 * CH; i += 256) {
            int r = i >> 7, c = i & (CH - 1);
            int gr = row0 + r;
            float xv = 0.0f, mv = 0.0f;
            if (gr < N) {
                xv = X[(size_t)gr * CH + c];
                mv = agg[(size_t)gr * CH + c] / fmaxf(deg[gr], 1.0f);
            }
            x_s[r][c] = xv;
            mean_s[r][c] = mv;
        }
    }
    __syncthreads();

    const int wave = tid >> 5;
    const int lane = tid & 31;
    const int col0 = wave * 16;
    const int m    = lane & 15;       // A row / B col / C-D col
    const int half = lane >> 4;       // 0: K(+0,+1), rows 0..7 ; 1: K(+2,+3), rows 8..15
    const int kb   = half * 2;

    v8f acc = {};   // zero-accumulate 16x16 f32 tile

    for (int k0 = 0; k0 < CH; k0 += 4) {
        v2f a0, b0, a1, b1;
        a0.x = mean_s[m][k0 + kb];
        a0.y = mean_s[m][k0 + kb + 1];
        b0.x = Wl[(size_t)(k0 + kb)     * CH + col0 + m];
        b0.y = Wl[(size_t)(k0 + kb + 1) * CH + col0 + m];
        acc = __builtin_amdgcn_wmma_f32_16x16x4_f32(false, a0, false, b0,
                                                    (short)0, acc, false, false);
        a1.x = x_s[m][k0 + kb];
        a1.y = x_s[m][k0 + kb + 1];
        b1.x = Wr[(size_t)(k0 + kb)     * CH + col0 + m];
        b1.y = Wr[(size_t)(k0 + kb + 1) * CH + col0 + m];
        acc = __builtin_amdgcn_wmma_f32_16x16x4_f32(false, a1, false, b1,
                                                    (short)0, acc, false, false);
    }

    const float bv = bias[col0 + m];
    if (full) {
        float* base = Y + (size_t)(row0 + half * 8) * CH + col0 + m;
        #pragma unroll
        for (int r = 0; r < 8; ++r) {
            float v = acc[r] + bv;
            v = (v > 0.0f) ? v : NEG_SLOPE * v;      // leaky_relu
            base[(size_t)r * CH] = v;
        }
    } else {
        #pragma unroll
        for (int r = 0; r < 8; ++r) {
            int grow = row0 + r + half * 8;
            if (grow < N) {
                float v = acc[r] + bv;
                v = (v > 0.0f) ? v : NEG_SLOPE * v;
                Y[(size_t)grow * CH + col0 + m] = v;
            }
        }
    }
}

// ---------- global max pool (segment_max over sorted batch) ----------
__global__ __launch_bounds__(256) void pool_max(const float* __restrict__ h,
                                                const int* __restrict__ batch,
                                                unsigned* __restrict__ pooled, int N) {
    long long idx = (long long)blockIdx.x * blockDim.x + threadIdx.x;
    if (idx >= (long long)N * CH) return;
    int i = (int)(idx >> 7), c = (int)idx & (CH - 1);
    int g = batch[i];
    atomicMax(&pooled[(size_t)g * CH + c], f2ord(h[idx]));
}

// ---------- final FC: out[g,o] = pooled[g,:] @ W_fc[:,o] + b_fc[o] ----------
__global__ void fc_kernel(const unsigned* __restrict__ pooled,
                          const float* __restrict__ Wfc,
                          const float* __restrict__ bfc,
                          float* __restrict__ out) {
    int tid = threadIdx.x;               // 128 threads: 64 graphs x 2 outputs
    int g = tid >> 1, o = tid & 1;
    float s = bfc[o];
    for (int k = 0; k < CH; ++k)
        s += ord2f(pooled[(size_t)g * CH + k]) * Wfc[k * 2 + o];
    out[g * 2 + o] = s;
}

extern "C" void kernel_launch(void* const* d_in, const int* in_sizes, int n_in,
                              void* d_out, int out_size, void* d_ws, size_t ws_size,
                              hipStream_t stream) {
    // inputs (setup_inputs order)
    const float* x     = (const float*)d_in[0];
    const int*   eidx  = (const int*)d_in[1];
    const int*   batch = (const int*)d_in[2];
    const float* W1l = (const float*)d_in[3];
    const float* b1  = (const float*)d_in[4];
    const float* W1r = (const float*)d_in[5];
    const float* W2l = (const float*)d_in[6];
    const float* b2  = (const float*)d_in[7];
    const float* W2r = (const float*)d_in[8];
    const float* W3l = (const float*)d_in[9];
    const float* b3  = (const float*)d_in[10];
    const float* W3r = (const float*)d_in[11];
    const float* Wfc = (const float*)d_in[12];
    const float* bfc = (const float*)d_in[13];

    const int N = in_sizes[0] / CH;
    const int E = in_sizes[1] / 2;
    const int* src = eidx;
    const int* dst = eidx + E;

    // workspace layout
    float* ws = (float*)d_ws;
    size_t NC = (size_t)N * CH;
    float* agg = ws;                 // [N,128]
    float* deg = agg + NC;           // [N]
    float* hA  = deg + N;            // [N,128]
    float* hB  = hA + NC;            // [N,128]
    unsigned* pooled = (unsigned*)(hB + NC);  // [64,128]

    const int TB = 256;
    const int zeroAggB = (int)((NC + TB - 1) / TB);
    const int zeroDegB = (N + TB - 1) / TB;
    const int edgeB    = (int)(((long long)E * 32 + TB - 1) / TB);
    const int gemmB    = (N + 15) / 16;
    const int poolB    = (int)((NC + TB - 1) / TB);

    // degree (layer-invariant)
    zero_f32<<<zeroDegB, TB, 0, stream>>>(deg, (size_t)N);
    degree_kernel<<<(E + TB - 1) / TB, TB, 0, stream>>>(dst, deg, E);

    // layer 1: x -> hA
    zero_f32<<<zeroAggB, TB, 0, stream>>>(agg, NC);
    scatter_add<<<edgeB, TB, 0, stream>>>(x, src, dst, agg, E);
    sage_gemm<<<gemmB, TB, 0, stream>>>(x, agg, deg, W1l, W1r, b1, hA, N);

    // layer 2: hA -> hB
    zero_f32<<<zeroAggB, TB, 0, stream>>>(agg, NC);
    scatter_add<<<edgeB, TB, 0, stream>>>(hA, src, dst, agg, E);
    sage_gemm<<<gemmB, TB, 0, stream>>>(hA, agg, deg, W2l, W2r, b2, hB, N);

    // layer 3: hB -> hA
    zero_f32<<<zeroAggB, TB, 0, stream>>>(agg, NC);
    scatter_add<<<edgeB, TB, 0, stream>>>(hB, src, dst, agg, E);
    sage_gemm<<<gemmB, TB, 0, stream>>>(hB, agg, deg, W3l, W3r, b3, hA, N);

    // pool + fc
    init_pooled<<<(NUM_GRAPHS * CH + TB - 1) / TB, TB, 0, stream>>>(pooled, NUM_GRAPHS * CH);
    pool_max<<<poolB, TB, 0, stream>>>(hA, batch, pooled, N);
    fc_kernel<<<1, NUM_GRAPHS * 2, 0, stream>>>(pooled, Wfc, bfc, (float*)d_out);
}